// NeuralTMT_71914932404431
// MI455X (gfx1250) — compile-verified
//
#include <hip/hip_runtime.h>

typedef float v2f __attribute__((ext_vector_type(2)));
typedef float v8f __attribute__((ext_vector_type(8)));

#define NEG_BIG_F (-4294967295.0f)   // -(2^32)+1, reference padding mask value

// 16 batches per block, 256 threads = 8 wave32s.
// LDS: x tile  : 64 rows (b_local*4+s) x 64 k, padded stride 68 -> 17.4 KB
//      D tiles : 8 chains x 16x16 f32                         ->  8.0 KB
__global__ __launch_bounds__(256, 2) void neuraltmt_fused(
    const float* __restrict__ IL, const float* __restrict__ LI,
    const float* __restrict__ UI, const float* __restrict__ IU,
    const float* __restrict__ alpha,
    const int* __restrict__ uid, const int* __restrict__ baskets,
    const int* __restrict__ iid, const int* __restrict__ neg_iid,
    float* __restrict__ out,
    int U, int V, int B, int L)
{
  constexpr int K  = 64;   // embedding dim (kIL == kUI == 64)
  constexpr int NB = 16;   // batches per block
  constexpr int XS = 68;   // padded x row stride in floats

  __shared__ float xlds[64 * XS];
  __shared__ float Dlds[8 * 256];

  const int tid  = threadIdx.x;
  const int wave = tid >> 5;
  const int lane = tid & 31;
  const int b0   = blockIdx.x * NB;
  const int nItems = V - 1;                 // padding index
  const float invL = 1.0f / (float)L;

  // ---------------- Stage 1: gather + mean-pool baskets -> x in LDS ----------
  // One wave per (b_local, s) pair; basket index is wave-uniform (scalar load),
  // 32 lanes cover the 64-float row with coalesced float2 loads.
  for (int pair = wave; pair < NB * 4; pair += 8) {
    const int bl = pair >> 2;
    const int s  = pair & 3;
    int b = b0 + bl; if (b >= B) b = B - 1;
    const long brow = ((long)s * B + b) * L;
    float ax = 0.f, ay = 0.f;
    for (int l = 0; l < L; ++l) {
      const int idx = baskets[brow + l];
      if (idx != nItems) {                  // pad_mask (LI[:, -1, :] is zero anyway)
        const float* r = LI + ((long)s * V + idx) * K + 2 * lane;
        v2f v = *(const v2f*)r;
        ax += v.x; ay += v.y;
      }
    }
    xlds[pair * XS + 2 * lane]     = ax * invL;   // mean divides by L (as reference)
    xlds[pair * XS + 2 * lane + 1] = ay * invL;
  }
  __syncthreads();

  // ---------------- Stage 2: fp32 WMMA 16x16x4 attention raw-dot tiles -------
  // Wave w -> batch-group g = w>>1 (4 batches), pn = w&1 (pos/neg).
  // A tile: M = (b4, p) rows of tgt = IL[p][iid/neg_iid];  B tile: N = (b4, s) rows of x.
  // D[m][n] with m=(bi,p), n=(bj,s); only bi==bj diagonal blocks are consumed.
  {
    const int g  = wave >> 1;
    const int pn = wave & 1;
    const int* sel = pn ? neg_iid : iid;
    const int mn = lane & 15;            // A-row index m / B-col index n
    const int kc = (lane >> 4) << 1;     // lanes 16-31 carry K+2,K+3
    const int p  = mn & 3;
    const int bi = mn >> 2;
    int bA = b0 + g * 4 + bi; if (bA >= B) bA = B - 1;
    const int aidx = sel[p * B + bA];
    const float* arow = IL + ((long)p * V + aidx) * K;
    const float* xrow = &xlds[(g * 16 + mn) * XS];

    float* D = &Dlds[wave * 256];
#if __has_builtin(__builtin_amdgcn_wmma_f32_16x16x4_f32)
    v8f acc = {0.f, 0.f, 0.f, 0.f, 0.f, 0.f, 0.f, 0.f};
#pragma unroll
    for (int k0 = 0; k0 < K; k0 += 4) {
      v2f av = *(const v2f*)(arow + k0 + kc);   // A: lanes 0-15 M rows, K=k0+kc..+1
      v2f bv = *(const v2f*)(xrow + k0 + kc);   // B: lanes 0-15 N cols, same K pair
      acc = __builtin_amdgcn_wmma_f32_16x16x4_f32(
          false, av, false, bv, (short)0, acc, false, false);
    }
#pragma unroll
    for (int i = 0; i < 8; ++i) {
      const int m = i + ((lane >> 4) << 3);     // C/D layout: VGPR i -> M=i / M=i+8
      D[m * 16 + mn] = acc[i];
    }
#else
    // VALU fallback (keeps kernel compiling if builtin is absent)
#pragma unroll
    for (int i = 0; i < 8; ++i) {
      const int m = i + ((lane >> 4) << 3);
      const int pm = m & 3, bm = m >> 2;
      int bb = b0 + g * 4 + bm; if (bb >= B) bb = B - 1;
      const float* ar = IL + ((long)pm * V + sel[pm * B + bb]) * K;
      const float* xr = &xlds[(g * 16 + mn) * XS];
      float d = 0.f;
      for (int k = 0; k < K; ++k) d += ar[k] * xr[k];
      D[m * 16 + mn] = d;
    }
#endif
  }
  __syncthreads();

  // ---------------- Stage 3: softmax-attend + MF dot + sigmoid blend ---------
  if (tid < 2 * 4 * NB) {                      // 128 outputs per block
    const int pn = tid & 1;                    // 0=pos, 1=neg
    const int j  = (tid >> 1) & 3;             // period
    const int bl = tid >> 3;                   // local batch 0..15
    int b = b0 + bl;
    const bool valid = (b < B);
    if (!valid) b = B - 1;
    const int g = bl >> 2, bi = bl & 3;

    const float* D = &Dlds[((g << 1) | pn) * 256 + (4 * bi + j) * 16 + 4 * bi];
    const float r0 = D[0], r1 = D[1], r2 = D[2], r3 = D[3];
    // w = raw/sqrt(64); exact zeros -> NEG_BIG before softmax (reference mask)
    const float w0 = (r0 == 0.f) ? NEG_BIG_F : r0 * 0.125f;
    const float w1 = (r1 == 0.f) ? NEG_BIG_F : r1 * 0.125f;
    const float w2 = (r2 == 0.f) ? NEG_BIG_F : r2 * 0.125f;
    const float w3 = (r3 == 0.f) ? NEG_BIG_F : r3 * 0.125f;
    const float mx = fmaxf(fmaxf(w0, w1), fmaxf(w2, w3));
    const float e0 = __expf(w0 - mx), e1 = __expf(w1 - mx);
    const float e2 = __expf(w2 - mx), e3 = __expf(w3 - mx);
    const float attn = (e0 * r0 + e1 * r1 + e2 * r2 + e3 * r3)
                     / (e0 + e1 + e2 + e3);    // sum_s softmax_s * raw_s

    const int* sel = pn ? neg_iid : iid;
    const int ii = sel[j * B + b];
    const float* urow = UI + ((long)j * U + uid[b]) * K;
    const float* irow = IU + (long)ii * K;
    float dot = 0.f;
#pragma unroll
    for (int k = 0; k < K; k += 4) {
      const float4 uu = *(const float4*)(urow + k);
      const float4 vv = *(const float4*)(irow + k);
      dot += uu.x * vv.x + uu.y * vv.y + uu.z * vv.z + uu.w * vv.w;
    }

    const float sg = 1.f / (1.f + __expf(-alpha[j]));   // sigmoid(alpha)
    if (valid)
      out[(long)((j << 1) | pn) * B + b] = sg * attn + (1.f - sg) * dot;
  }
}

extern "C" void kernel_launch(void* const* d_in, const int* in_sizes, int n_in,
                              void* d_out, int out_size, void* d_ws, size_t ws_size,
                              hipStream_t stream) {
  const float* IL      = (const float*)d_in[0];   // [4,V,64]
  const float* LI      = (const float*)d_in[1];   // [4,V,64]
  const float* UI      = (const float*)d_in[2];   // [4,U,64]
  const float* IU      = (const float*)d_in[3];   // [V,64]
  const float* alpha   = (const float*)d_in[4];   // [4]
  const int*   uid     = (const int*)d_in[5];     // [B]
  const int*   baskets = (const int*)d_in[6];     // [4,B,L]
  const int*   iid     = (const int*)d_in[7];     // [4,B]
  const int*   neg_iid = (const int*)d_in[8];     // [4,B]
  float*       out     = (float*)d_out;           // 8 x [B]

  const int V = in_sizes[0] / (4 * 64);
  const int U = in_sizes[2] / (4 * 64);
  const int B = in_sizes[5];
  const int L = in_sizes[6] / (4 * B);

  const int grid = (B + 15) / 16;
  hipLaunchKernelGGL(neuraltmt_fused, dim3(grid), dim3(256), 0, stream,
                     IL, LI, UI, IU, alpha, uid, baskets, iid, neg_iid, out,
                     U, V, B, L);
}